// CombinedProductLoss_85504208929330
// MI455X (gfx1250) — compile-verified
//
#include <hip/hip_runtime.h>
#include <math.h>

typedef __attribute__((ext_vector_type(2))) float v2f;
typedef __attribute__((ext_vector_type(8))) float v8f;
typedef __attribute__((ext_vector_type(4))) unsigned int u32x4;
typedef __attribute__((ext_vector_type(8))) int i32x8;
typedef __attribute__((ext_vector_type(4))) int i32x4;

#define B_ROWS 2048
#define D_DIM  128
#define C_COLS 50000
#define NUM_CT 391          // ceil(50000 / 128)
#define SB_STRIDE 68        // 64 k-values + 4 pad floats -> conflict-free ds_load_b64

#if __has_builtin(__builtin_amdgcn_tensor_load_to_lds)
#define USE_TDM 1
#else
#define USE_TDM 0
#endif

constexpr float ARC_SCALE = 64.0f;
constexpr float COS_M = 0.8775825618903728f;   // cos(0.5)
constexpr float SIN_M = 0.4794255386042030f;   // sin(0.5)
constexpr float TH_C  = -0.8775825618903728f;  // cos(pi - 0.5)
constexpr float MM_C  = 0.2397127693021015f;   // sin(pi - 0.5) * 0.5
constexpr float LS    = 0.1f;
constexpr float TRIP_MARGIN = 0.3f;
constexpr float BIGF  = 1e9f;

// ---------------------------------------------------------------------------
// Row L2-normalize: one wave per row of 128 floats (float4/lane).
// ---------------------------------------------------------------------------
__global__ __launch_bounds__(256)
void normalize_rows_kernel(const float* __restrict__ src,
                           float* __restrict__ dst,
                           float* __restrict__ sq_out,
                           int nrows)
{
    const int row  = blockIdx.x * 8 + (threadIdx.x >> 5);
    const int lane = threadIdx.x & 31;
    if (row >= nrows) return;

    const float4 v = *(const float4*)(src + (size_t)row * D_DIM + lane * 4);
    float ss = v.x * v.x + v.y * v.y + v.z * v.z + v.w * v.w;
    ss += __shfl_xor(ss, 1);
    ss += __shfl_xor(ss, 2);
    ss += __shfl_xor(ss, 4);
    ss += __shfl_xor(ss, 8);
    ss += __shfl_xor(ss, 16);

    const float inv = 1.0f / sqrtf(ss);
    float4 o;
    o.x = v.x * inv; o.y = v.y * inv; o.z = v.z * inv; o.w = v.w * inv;
    *(float4*)(dst + (size_t)row * D_DIM + lane * 4) = o;
    if (sq_out && lane == 0) sq_out[row] = ss;
}

// ---------------------------------------------------------------------------
// Fused ArcFace GEMM (V_WMMA_F32_16X16X4_F32).
// Workgroup = 256 threads = 8 waves; output tile = 128 rows x 128 cols.
// B half-panel (128 cols x 64 k) staged into LDS with stride-68 padding
// (conflict-free ds_load_b64) -- via the Tensor Data Mover when available
// (pad_enable: 4 DWORDs after every 64), else a manual cooperative copy.
// A streamed from global (L2 resident) via uniform base + 32-bit offset.
// ---------------------------------------------------------------------------
__global__ __launch_bounds__(256)
void arcface_gemm_kernel(const float* __restrict__ embn,
                         const float* __restrict__ wn,
                         const int*   __restrict__ labels,
                         float* __restrict__ s_label,
                         float* __restrict__ pse,
                         float* __restrict__ psum)
{
    __shared__ float sB[128 * SB_STRIDE];   // 34,816 bytes

    const int wave  = threadIdx.x >> 5;
    const int lane  = threadIdx.x & 31;
    const int l16   = lane & 15;
    const int khalf = lane >> 4;              // K pair {0,1} vs {2,3}
    const int ct      = blockIdx.x;
    const int colBase = ct * 128;
    const int rowBase = blockIdx.y * 128 + wave * 16;

    // A: lane reads emb_n[rowBase + l16][k + 2*khalf .. +1]; 32-bit offset.
    const unsigned aOff = (unsigned)(rowBase + l16) * D_DIM + 2 * khalf;

    v8f acc[8];
#pragma unroll
    for (int a = 0; a < 8; ++a) acc[a] = (v8f)(0.0f);

    for (int h = 0; h < 2; ++h) {
        __syncthreads();   // previous half fully consumed before restage

#if USE_TDM
        if (wave == 0) {
            // ---- Tensor DMA descriptor (D#): 2-D tile, zero-fill OOB rows ----
            const unsigned lds_addr = (unsigned)(size_t)(&sB[0]);   // flat low32 = LDS offset
            const unsigned long long ga =
                (unsigned long long)(size_t)wn +
                (((size_t)colBase * D_DIM + (size_t)h * 64) * sizeof(float));
            const unsigned tdim1 = (unsigned)(C_COLS - colBase);    // rows in tensor from tile start

            u32x4 g0;
            g0[0] = 1u;                                     // count=1, user mode, no gather
            g0[1] = lds_addr;                               // lds_addr[63:32]
            g0[2] = (unsigned)(ga & 0xffffffffu);           // global_addr[95:64]
            g0[3] = (unsigned)((ga >> 32) & 0x01ffffffu)    // global_addr[120:96]
                    | (2u << 30);                           // type = 2 ("image")

            i32x8 g1;
            g1[0] = (int)((2u << 16)      // data_size = 4 bytes
                        | (1u << 20)      // pad_enable
                        | (5u << 22)      // pad_interval: every 64 DWORDs
                        | (3u << 25));    // pad_amount: 4 DWORDs
            g1[1] = (int)(128u << 16);                      // tensor_dim0 = 128 (low16)
            g1[2] = (int)((tdim1 & 0xffffu) << 16);         // dim0 hi16=0 | tensor_dim1 lo16
            g1[3] = (int)((tdim1 >> 16) | (64u << 16));     // tensor_dim1 hi16 | tile_dim0=64
            g1[4] = 128;                                    // tile_dim1=128, tile_dim2=0
            g1[5] = 128;                                    // tensor_dim0_stride low32
            g1[6] = 0;                                      // stride hi | dim1_stride lo
            g1[7] = 0;

            i32x4 gz4 = {0, 0, 0, 0};
            i32x8 gz8 = {0, 0, 0, 0, 0, 0, 0, 0};
            __builtin_amdgcn_tensor_load_to_lds(g0, g1, gz4, gz4, gz8, 0);
            __builtin_amdgcn_s_wait_tensorcnt(0);
        }
        __syncthreads();
#else
        // Manual staging fallback: sB[c][kk] = w_n[colBase + c][h*64 + kk]
        for (int idx = threadIdx.x; idx < 128 * 16; idx += 256) {
            const int r  = idx >> 4;          // col within tile: 0..127
            const int c4 = idx & 15;          // float4 index within 64 k
            int col = colBase + r;
            int cc  = col < C_COLS ? col : (C_COLS - 1);
            const float4 v = *(const float4*)(wn + (size_t)cc * D_DIM + h * 64 + c4 * 4);
            *(float4*)(sB + r * SB_STRIDE + c4 * 4) = v;
        }
        __syncthreads();
#endif

#pragma unroll
        for (int k = 0; k < 64; k += 4) {
            const v2f af = *(const v2f*)(embn + aOff + h * 64 + k);
            v2f bf[8];
#pragma unroll
            for (int a = 0; a < 8; ++a)
                bf[a] = *(const v2f*)(sB + (16 * a + l16) * SB_STRIDE + k + 2 * khalf);
#pragma unroll
            for (int a = 0; a < 8; ++a)
                acc[a] = __builtin_amdgcn_wmma_f32_16x16x4_f32(
                    false, af, false, bf[a], (short)0, acc[a], false, false);
        }
    }

    // Epilogue: ArcFace margin + per-row partial reductions.
    // C layout: element v -> row = rowBase + v + 8*khalf, col = 16a + l16.
#pragma unroll
    for (int v = 0; v < 8; ++v) {
        const int row = rowBase + v + 8 * khalf;
        const int lbl = labels[row];
        float sum_exp = 0.0f;
        float sum_s   = 0.0f;
#pragma unroll
        for (int a = 0; a < 8; ++a) {
            const int col = colBase + 16 * a + l16;
            if (col < C_COLS) {
                const float cosv = acc[a][v];
                const float sinv = sqrtf(fminf(fmaxf(1.0f - cosv * cosv, 0.0f), 1.0f));
                float phi = cosv * COS_M - sinv * SIN_M;
                phi = (cosv > TH_C) ? phi : (cosv - MM_C);
                const bool is_lbl = (col == lbl);
                const float s = ARC_SCALE * (is_lbl ? phi : cosv);
                if (is_lbl) s_label[row] = s;   // unique writer per row
                sum_exp += __expf(s);
                sum_s   += s;
            }
        }
        sum_exp += __shfl_xor(sum_exp, 1);
        sum_exp += __shfl_xor(sum_exp, 2);
        sum_exp += __shfl_xor(sum_exp, 4);
        sum_exp += __shfl_xor(sum_exp, 8);
        sum_s   += __shfl_xor(sum_s, 1);
        sum_s   += __shfl_xor(sum_s, 2);
        sum_s   += __shfl_xor(sum_s, 4);
        sum_s   += __shfl_xor(sum_s, 8);
        if (l16 == 0) {
            pse [(size_t)ct * B_ROWS + row] = sum_exp;
            psum[(size_t)ct * B_ROWS + row] = sum_s;
        }
    }
}

// ---------------------------------------------------------------------------
// Batch-hard triplet: one workgroup per row b (emb stays L2-resident).
// ---------------------------------------------------------------------------
__global__ __launch_bounds__(256)
void triplet_kernel(const float* __restrict__ emb,
                    const float* __restrict__ embsq,
                    const int*   __restrict__ labels,
                    float* __restrict__ hp,
                    float* __restrict__ hn,
                    float* __restrict__ cnt)
{
    __shared__ float eb[D_DIM];
    __shared__ float smax[8], smin[8], scnt[8];

    const int b   = blockIdx.x;
    const int tid = threadIdx.x;
    if (tid < D_DIM) eb[tid] = emb[(size_t)b * D_DIM + tid];
    __syncthreads();

    const int   lb  = labels[b];
    const float sqb = embsq[b];

    float maxpos = 0.0f;        // matches (dist * pos_mask).max with no positives
    float minneg = 3.0e38f;
    float c      = 0.0f;

    for (int j = tid; j < B_ROWS; j += 256) {
        const float* ej = emb + (size_t)j * D_DIM;
        float dot = 0.0f;
#pragma unroll
        for (int k = 0; k < D_DIM; k += 4) {
            const float4 v = *(const float4*)(ej + k);
            dot += eb[k] * v.x + eb[k + 1] * v.y + eb[k + 2] * v.z + eb[k + 3] * v.w;
        }
        const float d = sqrtf(fmaxf(embsq[j] - 2.0f * dot + sqb, 0.0f) + 1e-16f);
        const bool pos = (labels[j] == lb) && (j != b);
        if (pos) { maxpos = fmaxf(maxpos, d); c += 1.0f; }
        const float nv = d + (pos ? BIGF : 0.0f) + ((j == b) ? BIGF : 0.0f);
        minneg = fminf(minneg, nv);
    }

    for (int m = 1; m < 32; m <<= 1) {
        maxpos = fmaxf(maxpos, __shfl_xor(maxpos, m));
        minneg = fminf(minneg, __shfl_xor(minneg, m));
        c      += __shfl_xor(c, m);
    }
    const int lane = tid & 31, w = tid >> 5;
    if (lane == 0) { smax[w] = maxpos; smin[w] = minneg; scnt[w] = c; }
    __syncthreads();
    if (tid == 0) {
        float M = smax[0], N = smin[0], Cc = scnt[0];
        for (int i = 1; i < 8; ++i) {
            M = fmaxf(M, smax[i]); N = fminf(N, smin[i]); Cc += scnt[i];
        }
        hp[b] = M; hn[b] = N; cnt[b] = Cc;
    }
}

// ---------------------------------------------------------------------------
// Final reduction: combine ArcFace partials + triplet rows into scalar loss.
// ---------------------------------------------------------------------------
__global__ __launch_bounds__(256)
void final_reduce_kernel(const float* __restrict__ pse,
                         const float* __restrict__ psum,
                         const float* __restrict__ slab,
                         const float* __restrict__ hp,
                         const float* __restrict__ hn,
                         const float* __restrict__ cnt,
                         float* __restrict__ out)
{
    __shared__ float sArc[8], sTri[8], sVal[8];
    const int tid = threadIdx.x;
    float arc = 0.0f, tri = 0.0f, val = 0.0f;

    for (int b = tid; b < B_ROWS; b += 256) {
        float se = 0.0f, ss = 0.0f;
        for (int ctile = 0; ctile < NUM_CT; ++ctile) {
            se += pse [(size_t)ctile * B_ROWS + b];
            ss += psum[(size_t)ctile * B_ROWS + b];
        }
        const float logZ = logf(se);
        arc += (1.0f - LS) * (logZ - slab[b]) + LS * (logZ - ss * (1.0f / (float)C_COLS));

        const float v = (cnt[b] > 0.0f) ? 1.0f : 0.0f;
        tri += fmaxf(hp[b] - hn[b] + TRIP_MARGIN, 0.0f) * v;
        val += v;
    }

    for (int m = 1; m < 32; m <<= 1) {
        arc += __shfl_xor(arc, m);
        tri += __shfl_xor(tri, m);
        val += __shfl_xor(val, m);
    }
    const int lane = tid & 31, w = tid >> 5;
    if (lane == 0) { sArc[w] = arc; sTri[w] = tri; sVal[w] = val; }
    __syncthreads();
    if (tid == 0) {
        float A = 0.0f, T = 0.0f, V = 0.0f;
        for (int i = 0; i < 8; ++i) { A += sArc[i]; T += sTri[i]; V += sVal[i]; }
        const float arc_mean = A / (float)B_ROWS;
        const float tri_loss = (V > 0.0f) ? (T / fmaxf(V, 1.0f)) : 0.0f;
        out[0] = arc_mean + 0.5f * tri_loss;
    }
}

// ---------------------------------------------------------------------------
extern "C" void kernel_launch(void* const* d_in, const int* in_sizes, int n_in,
                              void* d_out, int out_size, void* d_ws, size_t ws_size,
                              hipStream_t stream)
{
    const float* emb    = (const float*)d_in[0];   // [2048,128]
    const float* W      = (const float*)d_in[1];   // [50000,128]
    const int*   labels = (const int*)d_in[2];     // [2048]
    float* out = (float*)d_out;

    float* ws    = (float*)d_ws;
    float* wn    = ws;                                   // 50000*128
    float* embn  = wn    + (size_t)C_COLS * D_DIM;       // 2048*128
    float* embsq = embn  + (size_t)B_ROWS * D_DIM;       // 2048
    float* slab  = embsq + B_ROWS;                       // 2048
    float* pse   = slab  + B_ROWS;                       // 391*2048
    float* psum  = pse   + (size_t)NUM_CT * B_ROWS;      // 391*2048
    float* hp    = psum  + (size_t)NUM_CT * B_ROWS;      // 2048
    float* hn    = hp    + B_ROWS;                       // 2048
    float* cntp  = hn    + B_ROWS;                       // 2048

    normalize_rows_kernel<<<(C_COLS + 7) / 8, 256, 0, stream>>>(W, wn, nullptr, C_COLS);
    normalize_rows_kernel<<<(B_ROWS + 7) / 8, 256, 0, stream>>>(emb, embn, embsq, B_ROWS);

    dim3 grid(NUM_CT, B_ROWS / 128);
    arcface_gemm_kernel<<<grid, 256, 0, stream>>>(embn, wn, labels, slab, pse, psum);

    triplet_kernel<<<B_ROWS, 256, 0, stream>>>(emb, embsq, labels, hp, hn, cntp);

    final_reduce_kernel<<<1, 256, 0, stream>>>(pse, psum, slab, hp, hn, cntp, out);
}